// MultiBoxLoss_21638045237782
// MI455X (gfx1250) — compile-verified
//
#include <hip/hip_runtime.h>
#include <hip/hip_bf16.h>
#include <stdint.h>

#define Bb 64
#define Aa 8732
#define Gg 50
#define Cc 81
#define TILE 128
#define TB 69            // ceil(Aa/TILE)
#define NCH 8
#define CH 1092          // ceil(Aa/NCH)
#define NEG_POS_RATIO 3

typedef float    v2f __attribute__((ext_vector_type(2)));
typedef float    v8f __attribute__((ext_vector_type(8)));
typedef unsigned v4u __attribute__((ext_vector_type(4)));
typedef int      v4i __attribute__((ext_vector_type(4)));
typedef int      v8i __attribute__((ext_vector_type(8)));

// ---------------------------------------------------------------------------
// TDM: 1-D bulk copy of `ndwords` dwords global -> LDS via Tensor Data Mover.
// Descriptor built per CDNA5 ISA §8 (D# group0/group1 bitfields).
// This toolchain (clang-23 / therock-10.0) exposes the 6-arg builtin:
//   (v4u g0, v8i g1, v4i g2, v4i g3, v8i extra, i32 cpol)
// ---------------------------------------------------------------------------
#if __has_builtin(__builtin_amdgcn_tensor_load_to_lds)
#define HAVE_TDM 1
__device__ inline void tdm_load_1d(unsigned lds_addr, const void* gptr, unsigned ndwords) {
  unsigned long long ga = (unsigned long long)(uintptr_t)gptr;
  v4u g0;
  g0[0] = 1u;                                            // count=1 (valid), user mode
  g0[1] = lds_addr;                                      // lds_addr [63:32]
  g0[2] = (unsigned)(ga & 0xFFFFFFFFu);                  // global_addr low
  g0[3] = (unsigned)((ga >> 32) & 0x01FFFFFFu) | 0x80000000u; // addr hi + type=2
  v8i g1;
  g1[0] = (int)(2u << 16);                               // data_size = 4 bytes
  g1[1] = (int)((ndwords & 0xFFFFu) << 16);              // tensor_dim0[15:0]
  g1[2] = (int)(((ndwords >> 16) & 0xFFFFu) | (1u << 16)); // tensor_dim0 hi | tensor_dim1=1
  g1[3] = (int)((ndwords & 0xFFFFu) << 16);              // tile_dim0 = ndwords (<=65535)
  g1[4] = 1;                                             // tile_dim1=1, tile_dim2=0
  g1[5] = (int)ndwords;                                  // tensor_dim0_stride low
  g1[6] = 0;
  g1[7] = 0;
  v4i z4; z4[0] = 0; z4[1] = 0; z4[2] = 0; z4[3] = 0;    // groups 2/3 unused (<=2D)
  v8i z8; z8[0] = 0; z8[1] = 0; z8[2] = 0; z8[3] = 0;
  z8[4] = 0; z8[5] = 0; z8[6] = 0; z8[7] = 0;
  __builtin_amdgcn_tensor_load_to_lds(g0, g1, z4, z4, z8, 0);
}
#else
#define HAVE_TDM 0
#endif

// ---------------------------------------------------------------------------
// Block (256-thread) sum reduction. Result valid in thread 0.
// ---------------------------------------------------------------------------
__device__ inline float block_reduce_sum(float v, float* s_red) {
  for (int o = 16; o > 0; o >>= 1) v += __shfl_down(v, o, 32);
  const int w = threadIdx.x >> 5, l = threadIdx.x & 31;
  if (l == 0) s_red[w] = v;
  __syncthreads();
  float r = 0.f;
  if (threadIdx.x < 8) r = s_red[threadIdx.x];
  if (w == 0) {
    for (int o = 4; o > 0; o >>= 1) r += __shfl_down(r, o, 32);
  }
  __syncthreads();
  return r;
}

// ---------------------------------------------------------------------------
// Kernel 1: per (batch, 128-anchor tile): stage scores/gt via TDM, do IoU
// matching + smooth-L1 loc loss + log-softmax confidence (WMMA reduction).
// ---------------------------------------------------------------------------
__global__ __launch_bounds__(256) void k1_match_conf(
    const float* __restrict__ pred_boxes, const float* __restrict__ pred_scores,
    const float* __restrict__ gt_boxes, const int* __restrict__ gt_labels,
    float* __restrict__ conf, float* __restrict__ conf_neg,
    int* __restrict__ npp, float* __restrict__ partial_lc)
{
  __shared__ float s_scores[TILE * Cc];
  __shared__ float s_gtb[Gg * 4];
  __shared__ int   s_gtl[Gg];
  __shared__ int   s_tl[TILE];
  __shared__ int   s_pm[TILE];
  __shared__ float s_sum[TILE];
  __shared__ float s_red[8];

  const int b   = blockIdx.y;
  const int t   = blockIdx.x;
  const int a0  = t * TILE;
  const int n_a = min(TILE, Aa - a0);
  const int tid = threadIdx.x;

  const float* gscore = pred_scores + ((size_t)b * Aa + a0) * Cc;

  // ---- Phase 0: stage score tile + GT data into LDS ----
#if HAVE_TDM
  if (tid == 0) {
    tdm_load_1d((unsigned)(uintptr_t)s_scores, gscore, (unsigned)(n_a * Cc));
    tdm_load_1d((unsigned)(uintptr_t)s_gtb, gt_boxes + (size_t)b * Gg * 4, Gg * 4);
    tdm_load_1d((unsigned)(uintptr_t)s_gtl, gt_labels + (size_t)b * Gg, Gg);
    __builtin_amdgcn_s_wait_tensorcnt(0);
  }
#else
  for (int k = tid; k < n_a * Cc; k += 256) s_scores[k] = gscore[k];
  for (int k = tid; k < Gg * 4; k += 256) s_gtb[k] = gt_boxes[(size_t)b * Gg * 4 + k];
  for (int k = tid; k < Gg; k += 256) s_gtl[k] = gt_labels[(size_t)b * Gg + k];
#endif
  __syncthreads();

  // ---- Phase 1: IoU matching + smooth-L1 loc loss (one anchor / thread) ----
  float locp = 0.f;
  int   posc = 0;
  if (tid < TILE) {
    int tl = 0, pm = 0;
    if (tid < n_a) {
      const int a = a0 + tid;
      const float4 pb = reinterpret_cast<const float4*>(pred_boxes)[(size_t)b * Aa + a];
      float pbv[4] = {pb.x, pb.y, pb.z, pb.w};
      const float area_a = (pbv[2] - pbv[0]) * (pbv[3] - pbv[1]);
      float best = -1.f; int bi = 0;
      for (int g = 0; g < Gg; ++g) {
        const float gx0 = s_gtb[g * 4 + 0], gy0 = s_gtb[g * 4 + 1];
        const float gx1 = s_gtb[g * 4 + 2], gy1 = s_gtb[g * 4 + 3];
        const float iw = fminf(pbv[2], gx1) - fmaxf(pbv[0], gx0);
        const float ih = fminf(pbv[3], gy1) - fmaxf(pbv[1], gy0);
        const float inter = fmaxf(iw, 0.f) * fmaxf(ih, 0.f);
        const float area_b = (gx1 - gx0) * (gy1 - gy0);
        const float uni = fmaxf(area_a + area_b - inter, 1e-6f);
        const float iou = inter / uni;
        if (iou > best) { best = iou; bi = g; }   // first-max, like argmax
      }
      if (best > 0.5f) tl = s_gtl[bi];
      pm = (tl > 0) ? 1 : 0;
      if (pm) {
        float s = 0.f;
        #pragma unroll
        for (int cdim = 0; cdim < 4; ++cdim) {
          const float d = pbv[cdim] - s_gtb[bi * 4 + cdim];
          const float ad = fabsf(d);
          s += (ad < 1.f) ? 0.5f * d * d : ad - 0.5f;
        }
        locp = s; posc = 1;
      }
    }
    s_tl[tid] = tl;
    s_pm[tid] = pm;
  }
  __syncthreads();

  // ---- Phase 2: log-softmax confidence. 16 anchors per wave; lane layout
  // matches V_WMMA_F32_16X16X4_F32 A-matrix (M=lane%16, K-pair by lane half).
  const int wv = tid >> 5;
  const int L  = tid & 31;
  const int al = (wv << 4) + (L & 15);
  const int a_use = (al < n_a) ? al : 0;        // clamp tail lanes to valid row
  const float* row = s_scores + a_use * Cc;
  const int k0 = (L < 16) ? 0 : 2;

  float m0 = -3.4e38f, m1 = -3.4e38f;
  for (int c = k0; c < Cc; c += 4)     m0 = fmaxf(m0, row[c]);
  for (int c = k0 + 1; c < Cc; c += 4) m1 = fmaxf(m1, row[c]);
  float mm = fmaxf(m0, m1);
  mm = fmaxf(mm, __shfl_xor(mm, 16, 32));       // combine K-halves (same anchor)

  float e0 = 0.f, e1 = 0.f;
  for (int c = k0; c < Cc; c += 4)     e0 += __expf(row[c] - mm);
  for (int c = k0 + 1; c < Cc; c += 4) e1 += __expf(row[c] - mm);

#if __has_builtin(__builtin_amdgcn_wmma_f32_16x16x4_f32)
  // D = A(16x4 partial expsums) x B(4x16 ones): row m of D = sumexp(anchor m).
  v2f av; av[0] = e0; av[1] = e1;
  v2f bv; bv[0] = 1.f; bv[1] = 1.f;
  v8f cv = {};
  v8f dv = __builtin_amdgcn_wmma_f32_16x16x4_f32(false, av, false, bv,
                                                 (short)0, cv, false, false);
  // D layout: lane 0 regs r hold rows 0..7 (N=0); lane 16 regs hold rows 8..15.
  if (L == 0) {
    #pragma unroll
    for (int r = 0; r < 8; ++r) s_sum[(wv << 4) + r] = dv[r];
  } else if (L == 16) {
    #pragma unroll
    for (int r = 0; r < 8; ++r) s_sum[(wv << 4) + 8 + r] = dv[r];
  }
#else
  float et = e0 + e1;
  et += __shfl_xor(et, 16, 32);
  if (L < 16) s_sum[(wv << 4) + L] = et;
#endif
  __syncthreads();

  float cpos = 0.f;
  if (L < 16 && al < n_a) {
    const float sexp = s_sum[(wv << 4) + L];
    const int   tl = s_tl[al];
    const int   pm = s_pm[al];
    const float confv = mm + __logf(sexp) - s_scores[al * Cc + tl];
    const size_t gi = (size_t)b * Aa + a0 + al;
    conf[gi] = confv;
    conf_neg[gi] = pm ? -1.0f : confv;
    if (pm) cpos = confv;
  }

  const float lc = block_reduce_sum(locp + cpos, s_red);
  const float np = block_reduce_sum((float)posc, s_red);
  if (tid == 0) {
    partial_lc[b * TB + t] = lc;
    npp[b * TB + t] = (int)(np + 0.5f);
  }
}

// ---------------------------------------------------------------------------
// Kernel 2: exact hard-negative mining. Rank by counting (stable-tie by index)
// against the full per-batch conf_neg vector held in LDS (broadcast reads).
// ---------------------------------------------------------------------------
__global__ __launch_bounds__(256) void k2_hardneg(
    const float* __restrict__ conf, const float* __restrict__ conf_neg,
    const int* __restrict__ npp, float* __restrict__ partial_neg)
{
  __shared__ float s_conf[Aa];
  __shared__ float s_red[8];
  __shared__ int   s_np;
  const int b = blockIdx.y, c = blockIdx.x, tid = threadIdx.x;

  for (int j = tid; j < Aa; j += 256) s_conf[j] = conf_neg[(size_t)b * Aa + j];
  if (tid == 0) {
    int np = 0;
    for (int t = 0; t < TB; ++t) np += npp[b * TB + t];
    s_np = np;
  }
  __syncthreads();

  const int nn = min(NEG_POS_RATIO * s_np, Aa - 1);
  float acc = 0.f;
  const int i0 = c * CH;
  const int i1 = min(i0 + CH, Aa);
  for (int i = i0 + tid; i < i1; i += 256) {
    const float ci = s_conf[i];
    int cnt = 0;
    for (int j = 0; j < Aa; ++j) {
      const float cj = s_conf[j];
      cnt += (int)((cj > ci) || ((cj == ci) && (j < i)));
    }
    if (cnt < nn) acc += conf[(size_t)b * Aa + i];
  }
  const float r = block_reduce_sum(acc, s_red);
  if (tid == 0) partial_neg[b * NCH + c] = r;
}

// ---------------------------------------------------------------------------
// Kernel 3: deterministic final combine.
// ---------------------------------------------------------------------------
__global__ __launch_bounds__(256) void k3_final(
    const float* __restrict__ partial_lc, const float* __restrict__ partial_neg,
    const int* __restrict__ npp, float* __restrict__ out)
{
  __shared__ float s_red[8];
  const int tid = threadIdx.x;
  float acc = 0.f;
  for (int k = tid; k < Bb * TB; k += 256) acc += partial_lc[k];
  for (int k = tid; k < Bb * NCH; k += 256) acc += partial_neg[k];
  float den = 0.f;
  if (tid < Bb) {
    int np = 0;
    for (int t = 0; t < TB; ++t) np += npp[tid * TB + t];
    den = (float)max(np, 1);
  }
  const float num = block_reduce_sum(acc, s_red);
  const float d = block_reduce_sum(den, s_red);
  if (tid == 0) out[0] = num / d;
}

// ---------------------------------------------------------------------------
extern "C" void kernel_launch(void* const* d_in, const int* in_sizes, int n_in,
                              void* d_out, int out_size, void* d_ws, size_t ws_size,
                              hipStream_t stream)
{
  (void)in_sizes; (void)n_in; (void)out_size; (void)ws_size;
  const float* pred_boxes  = (const float*)d_in[0];
  const float* pred_scores = (const float*)d_in[1];
  const float* gt_boxes    = (const float*)d_in[2];
  const int*   gt_labels   = (const int*)d_in[3];

  char* ws = (char*)d_ws;
  size_t off = 0;
  float* conf        = (float*)(ws + off); off += (size_t)Bb * Aa * sizeof(float);
  float* conf_neg    = (float*)(ws + off); off += (size_t)Bb * Aa * sizeof(float);
  int*   npp         = (int*)  (ws + off); off += (size_t)Bb * TB * sizeof(int);
  float* partial_lc  = (float*)(ws + off); off += (size_t)Bb * TB * sizeof(float);
  float* partial_neg = (float*)(ws + off);

  k1_match_conf<<<dim3(TB, Bb), 256, 0, stream>>>(
      pred_boxes, pred_scores, gt_boxes, gt_labels,
      conf, conf_neg, npp, partial_lc);
  k2_hardneg<<<dim3(NCH, Bb), 256, 0, stream>>>(conf, conf_neg, npp, partial_neg);
  k3_final<<<1, 256, 0, stream>>>(partial_lc, partial_neg, npp, (float*)d_out);
}